// ChainLoss_56951266345693
// MI455X (gfx1250) — compile-verified
//
#include <hip/hip_runtime.h>
#include <cstdint>
#include <cstddef>

// Problem constants (fixed by the reference).
#define BB 32      // batch == wave32 lane count: lane b owns batch element b
#define TT 400     // time steps
#define PP 3500    // pdfs
#define SS 2000    // FST states
#define EE 50000   // FST arcs
#define NEGV (-1.0e30f)

#define NBLK 250   // persistent grid: 250 blocks x 8 waves = 2000 waves = SS
#define NTHR 256
#define WPB  8     // waves per block
#define SLOTS 64   // LDS arc slots per wave (avg in-degree ~25; overflow -> global)

#define SRC_BITS 11          // SS=2000 < 2048
#define SRC_MASK 0x7FF

// ---------------- CDNA5 async global->LDS copy (gfx1250) ----------------
__device__ __forceinline__ void async_copy_b32(void* lds_ptr, const void* gptr) {
    uint32_t lds_addr = (uint32_t)(uintptr_t)lds_ptr;
    asm volatile("global_load_async_to_lds_b32 %0, %1, off"
                 :: "v"(lds_addr), "v"(gptr)
                 : "memory");
}
__device__ __forceinline__ void wait_async0() {
    asm volatile("s_wait_asynccnt 0" ::: "memory");
}

// ---------------- device-scope alpha accessors ----------------
// alpha is the only cross-block-shared mutable data; dev-scoped accesses are
// coherent at L2, so the grid barrier needs NO cache writeback/invalidate.
__device__ __forceinline__ float a_load(const float* p) {
    return __hip_atomic_load((const float*)p, __ATOMIC_RELAXED,
                             __HIP_MEMORY_SCOPE_AGENT);
}
__device__ __forceinline__ void a_store(float* p, float v) {
    __hip_atomic_store(p, v, __ATOMIC_RELAXED, __HIP_MEMORY_SCOPE_AGENT);
}

// ---------------- lightweight software grid barrier ----------------
// bar[0] = arrive counter, bar[1] = generation.
__device__ __forceinline__ void grid_barrier(int* bar) {
    // Dev-scope stores ack from L2; drain this wave's before signaling.
    asm volatile("s_wait_storecnt 0x0" ::: "memory");
    __syncthreads();
    if (threadIdx.x == 0) {
        const int g = __hip_atomic_load(bar + 1, __ATOMIC_RELAXED,
                                        __HIP_MEMORY_SCOPE_AGENT);
        const int prev = __hip_atomic_fetch_add(bar, 1, __ATOMIC_ACQ_REL,
                                                __HIP_MEMORY_SCOPE_AGENT);
        if (prev == NBLK - 1) {
            __hip_atomic_store(bar, 0, __ATOMIC_RELAXED,
                               __HIP_MEMORY_SCOPE_AGENT);
            __hip_atomic_fetch_add(bar + 1, 1, __ATOMIC_RELEASE,
                                   __HIP_MEMORY_SCOPE_AGENT);
        } else {
            while (__hip_atomic_load(bar + 1, __ATOMIC_ACQUIRE,
                                     __HIP_MEMORY_SCOPE_AGENT) == g) {
                __builtin_amdgcn_s_sleep(1);
            }
        }
    }
    __syncthreads();
}

// ---------------- CSR build (deterministic, no atomics) ----------------
// One wave per destination state; 4x-unrolled coalesced sweep of dst[].
__global__ void k_cnt(const int* __restrict__ dst, int* __restrict__ cnt) {
    const int wave = threadIdx.x >> 5, lane = threadIdx.x & 31;
    const int s = blockIdx.x * WPB + wave;
    if (s >= SS) return;
    int c = 0, e = lane;
    for (; e + 96 < EE; e += 128) {
        const int d0 = dst[e], d1 = dst[e + 32], d2 = dst[e + 64], d3 = dst[e + 96];
        c += (d0 == s) + (d1 == s) + (d2 == s) + (d3 == s);
    }
    for (; e < EE; e += 32) c += (dst[e] == s);
    for (int o = 16; o > 0; o >>= 1) c += __shfl_down(c, o);
    if (lane == 0) cnt[s] = c;
}

// Exclusive scan of 2000 counts in one block (Hillis-Steele over 2048).
__global__ void k_scan(const int* __restrict__ cnt, int* __restrict__ off) {
    __shared__ int buf[2048];
    const int tid = threadIdx.x;                 // 1024 threads
    for (int i = tid; i < 2048; i += 1024) buf[i] = (i < SS) ? cnt[i] : 0;
    __syncthreads();
    for (int d = 1; d < 2048; d <<= 1) {
        const int i0 = tid, i1 = tid + 1024;
        int v0 = (i0 >= d) ? buf[i0 - d] : 0;
        int v1 = (i1 >= d) ? buf[i1 - d] : 0;
        __syncthreads();
        buf[i0] += v0;
        buf[i1] += v1;
        __syncthreads();
    }
    for (int i = tid; i < SS; i += 1024) off[i] = (i == 0) ? 0 : buf[i - 1];
    if (tid == 0) off[SS] = buf[SS - 1];
}

// Stable scatter into packed CSR via ballot+popcount; arc order preserved
// within each state -> bitwise deterministic downstream sums.
__global__ void k_fill(const int* __restrict__ src, const int* __restrict__ dst,
                       const int* __restrict__ pdf, const float* __restrict__ lw,
                       const int* __restrict__ off,
                       int* __restrict__ csr_sp, float* __restrict__ csr_lw) {
    const int wave = threadIdx.x >> 5, lane = threadIdx.x & 31;
    const int s = blockIdx.x * WPB + wave;
    if (s >= SS) return;
    int cursor = off[s];
    int e0 = 0;
    for (; e0 + 128 <= EE; e0 += 128) {
        const int d0 = dst[e0 + lane],      d1 = dst[e0 + 32 + lane];
        const int d2 = dst[e0 + 64 + lane], d3 = dst[e0 + 96 + lane];
        #define FILL_STEP(dk, eoff)                                            \
        {                                                                      \
            const unsigned mk = (unsigned)__ballot(dk == s);                   \
            if (dk == s) {                                                     \
                const int e   = e0 + (eoff) + lane;                            \
                const int pos = cursor + __popc(mk & ((1u << lane) - 1u));     \
                csr_sp[pos] = src[e] | (pdf[e] << SRC_BITS);                   \
                csr_lw[pos] = lw[e];                                           \
            }                                                                  \
            cursor += __popc(mk);                                              \
        }
        FILL_STEP(d0, 0) FILL_STEP(d1, 32) FILL_STEP(d2, 64) FILL_STEP(d3, 96)
        #undef FILL_STEP
    }
    for (; e0 < EE; e0 += 32) {
        const int e = e0 + lane;
        const bool match = (e < EE) && (dst[e] == s);
        const unsigned mk = (unsigned)__ballot(match);
        if (match) {
            const int pos = cursor + __popc(mk & ((1u << lane) - 1u));
            csr_sp[pos] = src[e] | (pdf[e] << SRC_BITS);
            csr_lw[pos] = lw[e];
        }
        cursor += __popc(mk);
    }
}

// ---------------- x transpose: [B,T,P] -> [T,P,B] ----------------
__global__ void k_xpose(const float* __restrict__ x, float* __restrict__ xtr) {
    __shared__ float tile[32][33];
    const int t  = blockIdx.y;
    const int p0 = blockIdx.x * 32;
    const int tx = threadIdx.x;               // 0..31
    const int ty = threadIdx.y;               // 0..7
    for (int r = ty; r < 32; r += 8) {        // r = batch b, tx = p offset
        const int p = p0 + tx;
        tile[r][tx] = (p < PP)
            ? x[(size_t)r * ((size_t)TT * PP) + (size_t)t * PP + p] : 0.0f;
    }
    __syncthreads();
    for (int r = ty; r < 32; r += 8) {        // r = p offset, tx = batch b
        const int p = p0 + r;
        if (p < PP) xtr[((size_t)t * PP + p) * BB + tx] = tile[tx][r];
    }
}

__global__ void k_barinit(int* __restrict__ bar) {
    if (threadIdx.x < 2) bar[threadIdx.x] = 0;
}

// ---------------- persistent forward recursion ----------------
template <bool XT>
__global__ void k_forward(const float* __restrict__ x,     // [B,T,P]
                          const float* __restrict__ xtr,   // [T,P,B] (if XT)
                          const int*   __restrict__ csr_sp,
                          const float* __restrict__ csr_lw,
                          const int*   __restrict__ off,
                          const float* __restrict__ ilp,
                          float* alphaA, float* alphaB,
                          int*   __restrict__ bar,
                          float* __restrict__ logtot,
                          float* __restrict__ out) {
    __shared__ int   ls_sp[WPB * SLOTS];
    __shared__ float ls_lw[WPB * SLOTS];
    __shared__ float red[NTHR];

    const int tid   = threadIdx.x;
    const int wave  = tid >> 5;
    const int lane  = tid & 31;                // batch index
    const int s     = blockIdx.x * WPB + wave; // owned state (grid == SS waves)
    const int lbase = wave * SLOTS;

    int a0 = 0, deg = 0, nlds = 0;
    if (s < SS) {
        a0   = off[s];
        deg  = off[s + 1] - a0;
        nlds = deg < SLOTS ? deg : SLOTS;
        // CDNA5 async staging: this wave's arc metadata -> LDS, once, reused 400x.
        for (int i = lane; i < nlds; i += 32) {
            async_copy_b32(&ls_sp[lbase + i], csr_sp + a0 + i);
            async_copy_b32(&ls_lw[lbase + i], csr_lw + a0 + i);
        }
        a_store(&alphaA[s * BB + lane], ilp[s]);   // alpha_0 broadcast
    }
    wait_async0();
    __syncthreads();

    grid_barrier(bar);                        // alpha_0 visible device-wide

    float* ain  = alphaA;
    float* aout = alphaB;

    for (int t = 0; t < TT; ++t) {
        if (s < SS) {
            const float* xt = XT ? (xtr + (size_t)t * ((size_t)PP * BB) + lane)
                                 : nullptr;
            const float* xl = XT ? nullptr
                                 : (x + (size_t)lane * ((size_t)TT * PP)
                                      + (size_t)t * PP);
            float m = -3.0e38f, sum = 0.0f;
            #pragma unroll 2
            for (int i = 0; i < nlds; ++i) {   // LDS fast path (always, in practice)
                const int   sp  = ls_sp[lbase + i];   // wave-uniform broadcast
                const float elw = ls_lw[lbase + i];
                const int e_src = sp & SRC_MASK;
                const int e_pdf = sp >> SRC_BITS;
                // alpha line + x_t line: both single coalesced 128B L2 hits.
                float v = a_load(&ain[e_src * BB + lane]) + elw;
                v += XT ? xt[e_pdf * BB] : xl[e_pdf];
                const float mn = fmaxf(m, v);  // streaming logsumexp
                sum = sum * __expf(m - mn) + __expf(v - mn);
                m = mn;
            }
            for (int i = nlds; i < deg; ++i) { // rare overflow: global CSR
                const int   sp  = csr_sp[a0 + i];
                const float elw = csr_lw[a0 + i];
                const int e_src = sp & SRC_MASK;
                const int e_pdf = sp >> SRC_BITS;
                float v = a_load(&ain[e_src * BB + lane]) + elw;
                v += XT ? xt[e_pdf * BB] : xl[e_pdf];
                const float mn = fmaxf(m, v);
                sum = sum * __expf(m - mn) + __expf(v - mn);
                m = mn;
            }
            a_store(&aout[s * BB + lane],
                    (m > -5.0e29f) ? (m + __logf(sum)) : NEGV);
        }
        grid_barrier(bar);
        float* tmp = ain; ain = aout; aout = tmp;
    }

    // ain == alpha_T. Blocks 0..31: per-batch logsumexp over states.
    if (blockIdx.x < BB) {
        const int b = blockIdx.x;
        float mm = -3.0e38f;
        for (int q = tid; q < SS; q += NTHR) mm = fmaxf(mm, a_load(&ain[q * BB + b]));
        red[tid] = mm; __syncthreads();
        for (int o = NTHR / 2; o > 0; o >>= 1) {
            if (tid < o) red[tid] = fmaxf(red[tid], red[tid + o]);
            __syncthreads();
        }
        mm = red[0]; __syncthreads();
        float acc = 0.0f;
        for (int q = tid; q < SS; q += NTHR)
            acc += __expf(a_load(&ain[q * BB + b]) - mm);
        red[tid] = acc; __syncthreads();
        for (int o = NTHR / 2; o > 0; o >>= 1) {
            if (tid < o) red[tid] += red[tid + o];
            __syncthreads();
        }
        if (tid == 0) a_store(&logtot[b], mm + __logf(red[0]));
    }
    grid_barrier(bar);
    if (blockIdx.x == 0 && wave == 0) {       // objf = mean_b logtot[b]
        float v = a_load(&logtot[lane]);
        for (int o = 16; o > 0; o >>= 1) v += __shfl_down(v, o);
        if (lane == 0) out[0] = v / (float)BB;
    }
}

extern "C" void kernel_launch(void* const* d_in, const int* in_sizes, int n_in,
                              void* d_out, int out_size, void* d_ws, size_t ws_size,
                              hipStream_t stream) {
    (void)in_sizes; (void)n_in; (void)out_size;
    const float* x   = (const float*)d_in[0];  // [B,T,P]
    const float* lw  = (const float*)d_in[1];  // [E]
    const float* ilp = (const float*)d_in[2];  // [S]
    const int*   src = (const int*)d_in[3];    // [E]
    const int*   dst = (const int*)d_in[4];    // [E]
    const int*   pdf = (const int*)d_in[5];    // [E]
    float* out = (float*)d_out;

    // Workspace carve-up (all re-initialized every call -> deterministic).
    char* w = (char*)d_ws;
    float* alphaA = (float*)w; w += (size_t)SS * BB * 4;
    float* alphaB = (float*)w; w += (size_t)SS * BB * 4;
    int*   csr_sp = (int*)w;   w += (size_t)EE * 4;
    float* csr_lw = (float*)w; w += (size_t)EE * 4;
    int*   off    = (int*)w;   w += (size_t)(SS + 1) * 4;
    int*   cnt    = (int*)w;   w += (size_t)SS * 4;
    float* logtot = (float*)w; w += (size_t)BB * 4;
    int*   bar    = (int*)w;   w += 2 * 4;
    uintptr_t xa = ((uintptr_t)w + 255) & ~(uintptr_t)255;
    float* xtr = (float*)xa;
    const size_t xtr_bytes = (size_t)TT * PP * BB * 4;  // ~179 MB
    const bool use_xt = ((uintptr_t)d_ws + ws_size) >= (xa + xtr_bytes);

    k_cnt    <<<NBLK, NTHR, 0, stream>>>(dst, cnt);
    k_scan   <<<1, 1024, 0, stream>>>(cnt, off);
    k_fill   <<<NBLK, NTHR, 0, stream>>>(src, dst, pdf, lw, off, csr_sp, csr_lw);
    k_barinit<<<1, 32, 0, stream>>>(bar);
    if (use_xt) {
        k_xpose<<<dim3((PP + 31) / 32, TT), dim3(32, 8), 0, stream>>>(x, xtr);
        k_forward<true ><<<NBLK, NTHR, 0, stream>>>(x, xtr, csr_sp, csr_lw, off,
                                                    ilp, alphaA, alphaB, bar,
                                                    logtot, out);
    } else {
        k_forward<false><<<NBLK, NTHR, 0, stream>>>(x, xtr, csr_sp, csr_lw, off,
                                                    ilp, alphaA, alphaB, bar,
                                                    logtot, out);
    }
}